// SAGE_57586921505022
// MI455X (gfx1250) — compile-verified
//
#include <hip/hip_runtime.h>

// ---------------------------------------------------------------------------
// GraphSAGE forward for MI455X (gfx1250, wave32).
// All dense 64x64 GEMMs run on V_WMMA_F32_16X16X4_F32 (full fp32 precision).
// Weights live in LDS in a pair-swizzled layout so every WMMA B-fragment is a
// single bank-conflict-free ds_load_b64. Segment-mean uses global_atomic_add_f32.
// ---------------------------------------------------------------------------

typedef __attribute__((ext_vector_type(2))) float v2f;
typedef __attribute__((ext_vector_type(8))) float v8f;

// Swizzled weight layout: element W[k][j] -> sW[(k>>1)*PSTRIDE + j*2 + (k&1)]
// B-fragment (W[k0][col], W[k0+1][col]) (k0 even) = one aligned b64 at
// sW + (k0>>1)*PSTRIDE + col*2. PSTRIDE=160 shifts consecutive pair-rows by
// 32 dwords (16 bank-pairs), making the 32-lane b64 fetch conflict-free.
#define PSTRIDE 160
#define SWZ_WORDS (32 * PSTRIDE)  // 5120 floats = 20 KB per 64x64 matrix

static __device__ __forceinline__ v8f wmma_f32x4(v2f a, v2f b, v8f c) {
  // 8 args: (neg_a, A, neg_b, B, c_mod, C, reuse_a, reuse_b)
  return __builtin_amdgcn_wmma_f32_16x16x4_f32(false, a, false, b, (short)0, c,
                                               false, false);
}

static __device__ __forceinline__ void load_swizzled(float* __restrict__ dst,
                                                     const float* __restrict__ src,
                                                     int t, int nthreads) {
  for (int i = t; i < 64 * 64; i += nthreads) {
    int k = i >> 6, j = i & 63;
    dst[(k >> 1) * PSTRIDE + j * 2 + (k & 1)] = src[i];
  }
}

// ---------------------------------------------------------------------------
__global__ __launch_bounds__(256) void zero_f32_kernel(float* __restrict__ p, long n) {
  long i = (long)blockIdx.x * blockDim.x + threadIdx.x;
  long stride = (long)gridDim.x * blockDim.x;
  for (; i < n; i += stride) p[i] = 0.0f;
}

// x[i,:] = emb[node_ids[i],:]   (float4 vectorized, 16 threads per row)
__global__ __launch_bounds__(256) void gather_emb_kernel(
    const float* __restrict__ emb, const int* __restrict__ ids,
    float* __restrict__ x, int n) {
  long tid = (long)blockIdx.x * blockDim.x + threadIdx.x;
  if (tid >= (long)n * 16) return;
  int i = (int)(tid >> 4), c = (int)(tid & 15);
  float4 v = *((const float4*)(emb + (long)ids[i] * 64) + c);
  *((float4*)(x + (long)i * 64) + c) = v;
}

// deg[dst[e]] += 1
__global__ __launch_bounds__(256) void degree_kernel(
    const int* __restrict__ dst, float* __restrict__ deg, int e) {
  int i = blockIdx.x * blockDim.x + threadIdx.x;
  if (i < e) atomicAdd(&deg[dst[i]], 1.0f);
}

// agg[dst[e],:] += X[src[e],:]   (16 threads per edge, float4 gather + 4 atomics)
__global__ __launch_bounds__(256) void scatter_kernel(
    const float* __restrict__ X, const int* __restrict__ src,
    const int* __restrict__ dst, float* __restrict__ agg, int e) {
  long tid = (long)blockIdx.x * blockDim.x + threadIdx.x;
  if (tid >= (long)e * 16) return;
  int eidx = (int)(tid >> 4), c = (int)(tid & 15);
  int s = src[eidx], d = dst[eidx];
  float4 v = *((const float4*)(X + (long)s * 64) + c);
  float* out = agg + (long)d * 64 + c * 4;
  atomicAdd(out + 0, v.x);
  atomicAdd(out + 1, v.y);
  atomicAdd(out + 2, v.z);
  atomicAdd(out + 3, v.w);
}

// ---------------------------------------------------------------------------
// OUT = act( X @ Wself + (AGG/max(deg,1)) @ Wneigh + bias )
// One wave = 16 node rows. 16 k-steps x 4 col-tiles x 2 (self+neigh) WMMAs.
// f32 WMMA layouts (wave32):
//   A 16x4:  lane L<16 row L, VGPR0/1 = K0/K1; lanes 16..31 = K2/K3
//   B 4x16:  lane L col (L&15), VGPR0/1 = rows K0/K1 (lanes<16) or K2/K3
//   C 16x16: lane L col (L&15), VGPR v = row v + 8*(L>>4)
// ---------------------------------------------------------------------------
__global__ __launch_bounds__(256) void sage_wmma_kernel(
    const float* __restrict__ X, const float* __restrict__ AGG,
    const float* __restrict__ DEG, const float* __restrict__ Wself,
    const float* __restrict__ Wneigh, const float* __restrict__ bias,
    float* __restrict__ OUT, int ntiles, int relu) {
  __shared__ float sWs[SWZ_WORDS];
  __shared__ float sWn[SWZ_WORDS];
  __shared__ float sB[64];
  int t = threadIdx.x;
  load_swizzled(sWs, Wself, t, 256);
  load_swizzled(sWn, Wneigh, t, 256);
  if (t < 64) sB[t] = bias[t];
  __syncthreads();

  int wave = t >> 5, lane = t & 31;
  int tileIdx = blockIdx.x * 8 + wave;
  if (tileIdx >= ntiles) return;  // wave-uniform exit (EXEC stays all-1 for WMMA)

  int base = tileIdx * 16;
  int r = lane & 15, khalf = lane >> 4;
  long row = base + r;
  const float* xr = X + row * 64;
  const float* ar = AGG + row * 64;
  float rdeg = 1.0f / fmaxf(DEG[row], 1.0f);
  int col2 = (lane & 15) * 2;

  v8f acc[4] = {};
#pragma unroll
  for (int kc = 0; kc < 16; ++kc) {
    int k0 = kc * 4 + khalf * 2;           // even
    int prow = (k0 >> 1) * PSTRIDE;        // swizzled pair-row base
    v2f as = *(const v2f*)(xr + k0);       // global_load_b64
    v2f an = *(const v2f*)(ar + k0) * rdeg;  // b64 + v_pk_mul_f32
#pragma unroll
    for (int jt = 0; jt < 4; ++jt) {
      v2f bs = *(const v2f*)(sWs + prow + jt * 32 + col2);  // ds_load_b64
      acc[jt] = wmma_f32x4(as, bs, acc[jt]);
      v2f bn = *(const v2f*)(sWn + prow + jt * 32 + col2);
      acc[jt] = wmma_f32x4(an, bn, acc[jt]);
    }
  }

#pragma unroll
  for (int jt = 0; jt < 4; ++jt) {
    int col = jt * 16 + r;
    float bv = sB[col];
#pragma unroll
    for (int v = 0; v < 8; ++v) {
      float val = acc[jt][v] + bv;
      if (relu) val = fmaxf(val, 0.0f);
      OUT[(long)(base + v + 8 * khalf) * 64 + col] = val;
    }
  }
}

// ---------------------------------------------------------------------------
// score[p] = MLP( H[src[p]] * H[dst[p]] )  -- relu(xW1+b1), relu(xW2+b2), xW3+b3
// One wave = 16 pairs; intermediates round-trip through a per-wave LDS tile to
// convert C-layout -> A-layout between WMMA layers.
// ---------------------------------------------------------------------------
__global__ __launch_bounds__(128) void mlp_score_kernel(
    const float* __restrict__ H, const int* __restrict__ src,
    const int* __restrict__ dst, const float* __restrict__ Wm1,
    const float* __restrict__ bm1, const float* __restrict__ Wm2,
    const float* __restrict__ bm2, const float* __restrict__ Wm3,
    const float* __restrict__ bm3, float* __restrict__ score, int npairs) {
  __shared__ float sW1[SWZ_WORDS];
  __shared__ float sW2[SWZ_WORDS];
  __shared__ float sW3[64];
  __shared__ float sb1[64];
  __shared__ float sb2[64];
  __shared__ float tile[4][16][68];  // row stride 272B (8B-aligned), padded

  int t = threadIdx.x;
  load_swizzled(sW1, Wm1, t, 128);
  load_swizzled(sW2, Wm2, t, 128);
  if (t < 64) {
    sW3[t] = Wm3[t];
    sb1[t] = bm1[t];
    sb2[t] = bm2[t];
  }
  __syncthreads();

  int wave = t >> 5, lane = t & 31;
  int tileIdx = blockIdx.x * 4 + wave;
  int base = tileIdx * 16;
  bool valid = (base < npairs);
  if (!valid) base = 0;  // keep all lanes live (EXEC all-1s for WMMA)

  int r = lane & 15, khalf = lane >> 4;
  int p = base + r;
  const float* hs = H + (long)src[p] * 64;
  const float* hd = H + (long)dst[p] * 64;
  int col2 = r * 2;

  // ---- layer 1: f = hs*hd elementwise, x1 = relu(f @ W1 + b1) ----
  v8f acc[4] = {};
#pragma unroll
  for (int kc = 0; kc < 16; ++kc) {
    int k0 = kc * 4 + khalf * 2;
    int prow = (k0 >> 1) * PSTRIDE;
    v2f a = *(const v2f*)(hs + k0) * *(const v2f*)(hd + k0);  // 2x b64 + pk_mul
#pragma unroll
    for (int jt = 0; jt < 4; ++jt) {
      v2f b = *(const v2f*)(sW1 + prow + jt * 32 + col2);
      acc[jt] = wmma_f32x4(a, b, acc[jt]);
    }
  }
#pragma unroll
  for (int jt = 0; jt < 4; ++jt) {
    int col = jt * 16 + r;
#pragma unroll
    for (int v = 0; v < 8; ++v)
      tile[wave][v + 8 * khalf][col] = fmaxf(acc[jt][v] + sb1[col], 0.0f);
  }
  asm volatile("s_wait_dscnt 0" ::: "memory");  // same-wave DS in-order; cross-lane RAW

  // ---- layer 2: x2 = relu(x1 @ W2 + b2) ----
  v8f acc2[4] = {};
#pragma unroll
  for (int kc = 0; kc < 16; ++kc) {
    int k0 = kc * 4 + khalf * 2;
    int prow = (k0 >> 1) * PSTRIDE;
    v2f a = *(const v2f*)&tile[wave][r][k0];  // ds_load_b64 (8B aligned)
#pragma unroll
    for (int jt = 0; jt < 4; ++jt) {
      v2f b = *(const v2f*)(sW2 + prow + jt * 32 + col2);
      acc2[jt] = wmma_f32x4(a, b, acc2[jt]);
    }
  }
#pragma unroll
  for (int jt = 0; jt < 4; ++jt) {
    int col = jt * 16 + r;
#pragma unroll
    for (int v = 0; v < 8; ++v)
      tile[wave][v + 8 * khalf][col] = fmaxf(acc2[jt][v] + sb2[col], 0.0f);
  }
  asm volatile("s_wait_dscnt 0" ::: "memory");

  // ---- layer 3: score = x2 @ W3 + b3 (64 -> 1), lanes 0..15 one row each ----
  if (valid && lane < 16) {
    float a = bm3[0];
#pragma unroll 8
    for (int k = 0; k < 64; ++k) a += tile[wave][lane][k] * sW3[k];
    score[base + lane] = a;
  }
}

// loss = mean(max(0, neg - pos + 1))
__global__ __launch_bounds__(256) void loss_kernel(
    const float* __restrict__ pos, const float* __restrict__ neg,
    float* __restrict__ out, int p) {
  __shared__ float red[256];
  int i = blockIdx.x * 256 + threadIdx.x;
  float v = 0.0f;
  if (i < p) v = fmaxf(neg[i] - pos[i] + 1.0f, 0.0f) / (float)p;
  red[threadIdx.x] = v;
  __syncthreads();
  for (int s = 128; s > 0; s >>= 1) {
    if (threadIdx.x < s) red[threadIdx.x] += red[threadIdx.x + s];
    __syncthreads();
  }
  if (threadIdx.x == 0) atomicAdd(out, red[0]);
}

// ---------------------------------------------------------------------------
extern "C" void kernel_launch(void* const* d_in, const int* in_sizes, int n_in,
                              void* d_out, int out_size, void* d_ws,
                              size_t ws_size, hipStream_t stream) {
  const int* node_ids = (const int*)d_in[0];
  const int* edge_src = (const int*)d_in[1];
  const int* edge_dst = (const int*)d_in[2];
  const int* pos_src = (const int*)d_in[3];
  const int* pos_dst = (const int*)d_in[4];
  const int* neg_src = (const int*)d_in[5];
  const int* neg_dst = (const int*)d_in[6];
  const float* emb = (const float*)d_in[7];
  const float* W_self1 = (const float*)d_in[8];
  const float* W_neigh1 = (const float*)d_in[9];
  const float* b1 = (const float*)d_in[10];
  const float* W_self2 = (const float*)d_in[11];
  const float* W_neigh2 = (const float*)d_in[12];
  const float* b2 = (const float*)d_in[13];
  const float* Wm1 = (const float*)d_in[14];
  const float* bm1 = (const float*)d_in[15];
  const float* Wm2 = (const float*)d_in[16];
  const float* bm2 = (const float*)d_in[17];
  const float* Wm3 = (const float*)d_in[18];
  const float* bm3 = (const float*)d_in[19];

  const int N = in_sizes[0];  // 100000 nodes
  const int E = in_sizes[1];  // 1600000 edges
  const int P = in_sizes[3];  // 100000 pairs

  float* ws = (float*)d_ws;
  float* x = ws;                      // N*64
  float* agg = x + (size_t)N * 64;    // N*64
  float* h1 = agg + (size_t)N * 64;   // N*64
  float* deg = h1 + (size_t)N * 64;   // N
  float* pos_s = deg + N;             // P
  float* neg_s = pos_s + P;           // P

  float* out = (float*)d_out;
  float* h2 = out + 1;  // h written directly into d_out[1..]

  const int ntiles = (N + 15) / 16;  // N % 16 == 0 for this problem
  const int ptiles = (P + 15) / 16;

  // zero scratch + loss accumulator
  zero_f32_kernel<<<1024, 256, 0, stream>>>(agg, (long)N * 64);
  zero_f32_kernel<<<64, 256, 0, stream>>>(deg, N);
  zero_f32_kernel<<<1, 256, 0, stream>>>(out, 1);

  // x = emb[node_ids]
  {
    long nt = (long)N * 16;
    gather_emb_kernel<<<(int)((nt + 255) / 256), 256, 0, stream>>>(emb, node_ids, x, N);
  }
  // degrees (shared by both layers)
  degree_kernel<<<(E + 255) / 256, 256, 0, stream>>>(edge_dst, deg, E);

  // layer 1: aggregate + GEMM(+relu)
  {
    long nt = (long)E * 16;
    scatter_kernel<<<(int)((nt + 255) / 256), 256, 0, stream>>>(x, edge_src, edge_dst, agg, E);
  }
  sage_wmma_kernel<<<(ntiles + 7) / 8, 256, 0, stream>>>(
      x, agg, deg, W_self1, W_neigh1, b1, h1, ntiles, /*relu=*/1);

  // layer 2: re-aggregate on h1 + GEMM (no relu) -> h2 (== d_out[1..])
  zero_f32_kernel<<<1024, 256, 0, stream>>>(agg, (long)N * 64);
  {
    long nt = (long)E * 16;
    scatter_kernel<<<(int)((nt + 255) / 256), 256, 0, stream>>>(h1, edge_src, edge_dst, agg, E);
  }
  sage_wmma_kernel<<<(ntiles + 7) / 8, 256, 0, stream>>>(
      h1, agg, deg, W_self2, W_neigh2, b2, h2, ntiles, /*relu=*/0);

  // pair scoring MLP (WMMA) for pos & neg
  mlp_score_kernel<<<(ptiles + 3) / 4, 128, 0, stream>>>(
      h2, pos_src, pos_dst, Wm1, bm1, Wm2, bm2, Wm3, bm3, pos_s, P);
  mlp_score_kernel<<<(ptiles + 3) / 4, 128, 0, stream>>>(
      h2, neg_src, neg_dst, Wm1, bm1, Wm2, bm2, Wm3, bm3, neg_s, P);

  // margin loss -> d_out[0]
  loss_kernel<<<(P + 255) / 256, 256, 0, stream>>>(pos_s, neg_s, out, P);
}